// ABC_2D_84421877170803
// MI455X (gfx1250) — compile-verified
//
#include <hip/hip_runtime.h>

typedef __attribute__((ext_vector_type(2))) float v2f;
typedef __attribute__((ext_vector_type(8))) float v8f;

#define P_PIX 16384
#define KNUM  16
#define CKD   144
#define IMGSZ (16 * 128 * 128)   // C*H*W = 262144
#define NBATCH 8

// One wave32 per pixel.
//   A (SRC0) = weights[p]: 16(KN) x 4 f32 per chunk, 36 chunks (K=144)
//   B (SRC1) = gathered patches: 4 x 16 (cols = batch, cols 8..15 duplicate 0..7)
//   D        = 16(KN) x 16(batch) f32, store only cols 0..7
__global__ __launch_bounds__(256)
void hashconv_wmma_f32(const float* __restrict__ x,
                       const float* __restrict__ w,
                       const long long* __restrict__ hidx,
                       float* __restrict__ out)
{
    const int lane  = threadIdx.x & 31;
    const int wave  = threadIdx.x >> 5;
    const int khalf = lane >> 4;     // which K-half of the 4-wide chunk this lane holds
    const int nrow  = lane & 15;     // A: matrix row (KN); B/D: column (batch)

    // pixel id, forced wave-uniform so hash_idx loads become scalar s_loads
    const int p = __builtin_amdgcn_readfirstlane((int)(blockIdx.x * 8u + wave));

    // A source: row `nrow` of weights[p]; lane's K offset inside each 4-chunk is 2*khalf
    const float* Ap = w + ((size_t)p * KNUM + nrow) * CKD + 2 * khalf;
    // wave-uniform hash row for this pixel
    const long long* Hp = hidx + (size_t)p * CKD;
    // 32-bit element offset for this lane's batch column off the *uniform* x base
    // (cols 8..15 alias batches 0..7 -> identical addresses, cache-merged, never stored)
    const unsigned boff = (unsigned)(nrow & 7) * (unsigned)IMGSZ;

    v8f acc0 = {0.f, 0.f, 0.f, 0.f, 0.f, 0.f, 0.f, 0.f};
    v8f acc1 = {0.f, 0.f, 0.f, 0.f, 0.f, 0.f, 0.f, 0.f};

#pragma unroll 12
    for (int c = 0; c < CKD / 4; ++c) {
        // A chunk: K = 4c + {0,1} (lanes 0-15) / 4c + {2,3} (lanes 16-31), one b64 load
        v2f a = *(const v2f*)(Ap + 4 * c);

        // scalar loads of the 4 int64 hash indices for this chunk (values < 2^18)
        int j0 = (int)Hp[4 * c + 0];
        int j1 = (int)Hp[4 * c + 1];
        int j2 = (int)Hp[4 * c + 2];
        int j3 = (int)Hp[4 * c + 3];

        // per-lane K-half select + 32-bit offset -> global_load saddr+voffset form
        unsigned k0 = (unsigned)(khalf ? j2 : j0) + boff;
        unsigned k1 = (unsigned)(khalf ? j3 : j1) + boff;

        // B chunk: per-lane gathers from L2-resident x
        v2f b;
        b.x = x[k0];
        b.y = x[k1];

        if (c & 1)
            acc1 = __builtin_amdgcn_wmma_f32_16x16x4_f32(
                false, a, false, b, (short)0, acc1, false, false);
        else
            acc0 = __builtin_amdgcn_wmma_f32_16x16x4_f32(
                false, a, false, b, (short)0, acc0, false, false);
    }

    v8f acc = acc0 + acc1;

    // D: VGPR r holds (M = r + 8*khalf, N = nrow); out[b, k, p] with b=N, k=M
    if (nrow < NBATCH) {
        float* op = out + (size_t)(nrow * KNUM + 8 * khalf) * P_PIX + p;
#pragma unroll
        for (int r = 0; r < 8; ++r)
            op[(size_t)r * P_PIX] = acc[r];
    }
}

extern "C" void kernel_launch(void* const* d_in, const int* in_sizes, int n_in,
                              void* d_out, int out_size, void* d_ws, size_t ws_size,
                              hipStream_t stream) {
    (void)in_sizes; (void)n_in; (void)d_ws; (void)ws_size; (void)out_size;
    const float*     x = (const float*)d_in[0];
    const float*     w = (const float*)d_in[1];
    const long long* h = (const long long*)d_in[2];
    float*           o = (float*)d_out;

    // 16384 pixels, 8 waves (256 threads) per block -> 2048 blocks
    hashconv_wmma_f32<<<P_PIX / 8, 256, 0, stream>>>(x, w, h, o);
}